// SparseConcatenate_80161269612699
// MI455X (gfx1250) — compile-verified
//
#include <hip/hip_runtime.h>
#include <hip/hip_bf16.h>
#include <stdint.h>

// ---------------------------------------------------------------------------
// SparseConcatenate for MI455X (gfx1250).
// Dedup via 16M-bit presence bitmap + popcount ranking (keys are 24-bit since
// coords < 256 and lexicographic row order == packed-key integer order).
// Feature assembly: one single coalesced pass over the 205MB output, with the
// irregular 256B-row gathers done by the Tensor Data Mover (gather mode,
// 32-bit row indices) into LDS, overlapped with rank/index work.
// ---------------------------------------------------------------------------

typedef __attribute__((ext_vector_type(4))) unsigned int v4u;
typedef __attribute__((ext_vector_type(8))) int          v8i;
typedef __attribute__((ext_vector_type(4))) int          v4i;

#define KEY_BITS        24u
#define NBITS           (1u << KEY_BITS)          // 16,777,216 possible keys
#define NWORDS          (NBITS / 32u)             // 524,288 bitmap words (2MB)
#define NCHUNK          512u
#define WORDS_PER_CHUNK (NWORDS / NCHUNK)         // 1024

__device__ __forceinline__ unsigned pack_key(const int* __restrict__ loc, int row) {
    unsigned x = (unsigned)loc[row * 3 + 0];
    unsigned y = (unsigned)loc[row * 3 + 1];
    unsigned z = (unsigned)loc[row * 3 + 2];
    return (x << 16) | (y << 8) | z;
}

__device__ __forceinline__ unsigned rank_of(unsigned key,
                                            const unsigned* __restrict__ bitmap,
                                            const unsigned* __restrict__ wp) {
    unsigned w = key >> 5, b = key & 31u;
    return wp[w] + __popc(bitmap[w] & ((1u << b) - 1u));
}

// ---- small utility kernels -------------------------------------------------

__global__ void k_fill_u32(unsigned* __restrict__ p, unsigned v, int n) {
    int i = blockIdx.x * blockDim.x + threadIdx.x;
    if (i < n) p[i] = v;
}

__global__ void k_bitmap(const int* __restrict__ aloc, const int* __restrict__ bloc,
                         unsigned* __restrict__ bitmap, int Na, int Nb) {
    int i = blockIdx.x * blockDim.x + threadIdx.x;
    if (i >= Na + Nb) return;
    unsigned key = (i < Na) ? pack_key(aloc, i) : pack_key(bloc, i - Na);
    atomicOr(&bitmap[key >> 5], 1u << (key & 31u));
}

__global__ void k_chunksums(const unsigned* __restrict__ bitmap, unsigned* __restrict__ sums) {
    int t = blockIdx.x * blockDim.x + threadIdx.x;
    if (t >= (int)NCHUNK) return;
    const unsigned* p = bitmap + (size_t)t * WORDS_PER_CHUNK;
    unsigned s = 0;
    for (int i = 0; i < (int)WORDS_PER_CHUNK; ++i) s += __popc(p[i]);
    sums[t] = s;
}

__global__ void k_scan(const unsigned* __restrict__ sums, unsigned* __restrict__ offs,
                       unsigned* __restrict__ total) {
    __shared__ unsigned s[NCHUNK];
    int t = threadIdx.x;                    // launched with NCHUNK threads
    unsigned mine = sums[t];
    s[t] = mine;
    __syncthreads();
    for (int off = 1; off < (int)NCHUNK; off <<= 1) {
        unsigned v = (t >= off) ? s[t - off] : 0u;
        __syncthreads();
        s[t] += v;
        __syncthreads();
    }
    offs[t] = s[t] - mine;                  // exclusive prefix
    if (t == (int)NCHUNK - 1) *total = s[t];
}

__global__ void k_wordprefix(const unsigned* __restrict__ bitmap,
                             const unsigned* __restrict__ offs,
                             unsigned* __restrict__ wp) {
    int t = blockIdx.x * blockDim.x + threadIdx.x;
    if (t >= (int)NCHUNK) return;
    unsigned run = offs[t];
    size_t base = (size_t)t * WORDS_PER_CHUNK;
    for (int i = 0; i < (int)WORDS_PER_CHUNK; ++i) {
        wp[base + i] = run;
        run += __popc(bitmap[base + i]);
    }
}

__global__ void k_inverse(const int* __restrict__ aloc, const int* __restrict__ bloc,
                          const unsigned* __restrict__ bitmap, const unsigned* __restrict__ wp,
                          int* __restrict__ invA, int* __restrict__ invB, int Na, int Nb) {
    int i = blockIdx.x * blockDim.x + threadIdx.x;
    if (i >= Na + Nb) return;
    if (i < Na) {
        invA[rank_of(pack_key(aloc, i), bitmap, wp)] = i;
    } else {
        invB[rank_of(pack_key(bloc, i - Na), bitmap, wp)] = i - Na;
    }
}

__global__ void k_fill_loc(const unsigned* __restrict__ bitmap,
                           const unsigned* __restrict__ wp, int* __restrict__ outLoc) {
    int w = blockIdx.x * blockDim.x + threadIdx.x;
    if (w >= (int)NWORDS) return;
    unsigned bits = bitmap[w];
    unsigned base = wp[w];
    while (bits) {
        unsigned b = __ffs(bits) - 1u;
        unsigned key = ((unsigned)w << 5) | b;
        size_t o = (size_t)base * 3;
        outLoc[o + 0] = (int)(key >> 16);
        outLoc[o + 1] = (int)((key >> 8) & 255u);
        outLoc[o + 2] = (int)(key & 255u);
        ++base;
        bits &= bits - 1u;
    }
}

__global__ void k_fill_tail(int* __restrict__ outLoc, const unsigned* __restrict__ total, int N) {
    int i = blockIdx.x * blockDim.x + threadIdx.x;
    if (i >= N) return;
    if ((unsigned)i >= *total) {
        outLoc[i * 3 + 0] = -1;
        outLoc[i * 3 + 1] = -1;
        outLoc[i * 3 + 2] = -1;
    }
}

// ---- feature assembly: TDM gather (Global->LDS), coalesced store -----------
// Each wave handles 8 output rows. Two TDM gather-mode descriptors (A half,
// B half) pull 8 scattered 256B feature rows into LDS asynchronously; after
// s_wait_tensorcnt the wave writes 8 contiguous 512B output rows (float4/lane).

__device__ __forceinline__ v4u make_g0_gather(unsigned ldsAddr, uint64_t gaddr) {
    v4u g0;
    g0.x = 1u | (1u << 30) | (1u << 31);   // count=1 valid | 32-bit indices | gather_mode
    g0.y = ldsAddr;                         // D#.lds_addr (bytes)
    g0.z = (unsigned)(gaddr & 0xFFFFFFFFull);
    g0.w = (unsigned)((gaddr >> 32) & 0x01FFFFFFull) | (2u << 30);  // addr[56:32] | type=2
    return g0;
}

__device__ __forceinline__ v8i make_g1_gather(unsigned tensorRows) {
    v8i g1;
    g1[0] = (int)(2u << 16);                              // data_size=2 (4 bytes)
    g1[1] = (int)(64u << 16);                             // tensor_dim0[15:0]=64
    g1[2] = (int)((tensorRows & 0xFFFFu) << 16);          // tensor_dim1[15:0]
    g1[3] = (int)(((tensorRows >> 16) & 0xFFFFu) | (64u << 16)); // tensor_dim1 hi | tile_dim0=64
    g1[4] = (int)8u;                                      // tile_dim1 = 8 valid indices
    g1[5] = 64;                                           // tensor_dim0_stride[31:0] = 64 elems
    g1[6] = 0;
    g1[7] = 0;
    return g1;
}

__global__ void k_gather_features(const float* __restrict__ afeat, const float* __restrict__ bfeat,
                                  const int* __restrict__ invA, const int* __restrict__ invB,
                                  float* __restrict__ outFeat, int N, int Na, int Nb) {
    __shared__ float tile[8][2][8][64];   // [wave][src][row][col]  = 32 KB
    const int lane  = threadIdx.x & 31;
    const int wave  = threadIdx.x >> 5;
    const int uStart = (blockIdx.x * 8 + wave) * 8;
    if (uStart >= N) return;

    int ia = -1, ib = -1;
    if (lane < 8 && (uStart + lane) < N) {
        ia = invA[uStart + lane];
        ib = invB[uStart + lane];
    }
    int ca = ia < 0 ? 0 : ia;   // clamped gather index (row 0 if unused)
    int cb = ib < 0 ? 0 : ib;

    // Row indices into descriptor groups 2/3 (32-bit index mode: 4 per group).
    v4i g2a, g3a, g2b, g3b;
    g2a[0] = __shfl(ca, 0, 32); g2a[1] = __shfl(ca, 1, 32);
    g2a[2] = __shfl(ca, 2, 32); g2a[3] = __shfl(ca, 3, 32);
    g3a[0] = __shfl(ca, 4, 32); g3a[1] = __shfl(ca, 5, 32);
    g3a[2] = __shfl(ca, 6, 32); g3a[3] = __shfl(ca, 7, 32);
    g2b[0] = __shfl(cb, 0, 32); g2b[1] = __shfl(cb, 1, 32);
    g2b[2] = __shfl(cb, 2, 32); g2b[3] = __shfl(cb, 3, 32);
    g3b[0] = __shfl(cb, 4, 32); g3b[1] = __shfl(cb, 5, 32);
    g3b[2] = __shfl(cb, 6, 32); g3b[3] = __shfl(cb, 7, 32);

    unsigned ldsA = (unsigned)(uintptr_t)&tile[wave][0][0][0];
    unsigned ldsB = (unsigned)(uintptr_t)&tile[wave][1][0][0];
    v4u g0a = make_g0_gather(ldsA, (uint64_t)(uintptr_t)afeat);
    v4u g0b = make_g0_gather(ldsB, (uint64_t)(uintptr_t)bfeat);
    v8i g1a = make_g1_gather((unsigned)Na);
    v8i g1b = make_g1_gather((unsigned)Nb);

    v8i g4 = {0, 0, 0, 0, 0, 0, 0, 0};    // extra operand of the 6-arg builtin

    // Async TDM gathers: 8 rows x 64 f32 each, scattered source rows -> LDS.
    __builtin_amdgcn_tensor_load_to_lds(g0a, g1a, g2a, g3a, g4, 0);
    __builtin_amdgcn_tensor_load_to_lds(g0b, g1b, g2b, g3b, g4, 0);
    __builtin_amdgcn_s_wait_tensorcnt(0);
    __asm__ volatile("" ::: "memory");   // LDS now holds TDM data

    for (int r = 0; r < 8; ++r) {
        int u = uStart + r;
        if (u >= N) break;
        int srcA = __shfl(ia, r, 32);
        int srcB = __shfl(ib, r, 32);
        float4 v;
        if (lane < 16) {
            v = (srcA >= 0) ? *(const float4*)&tile[wave][0][r][lane * 4]
                            : make_float4(0.f, 0.f, 0.f, 0.f);
        } else {
            v = (srcB >= 0) ? *(const float4*)&tile[wave][1][r][(lane - 16) * 4]
                            : make_float4(0.f, 0.f, 0.f, 0.f);
        }
        *(float4*)&outFeat[(size_t)u * 128 + lane * 4] = v;   // coalesced 512B row
    }
}

// ---------------------------------------------------------------------------

extern "C" void kernel_launch(void* const* d_in, const int* in_sizes, int n_in,
                              void* d_out, int out_size, void* d_ws, size_t ws_size,
                              hipStream_t stream) {
    const int*   aloc  = (const int*)d_in[0];
    const float* afeat = (const float*)d_in[1];
    const int*   bloc  = (const int*)d_in[2];
    const float* bfeat = (const float*)d_in[3];

    const int Na = in_sizes[0] / 3;          // 200000
    const int Nb = in_sizes[2] / 3;          // 200000
    const int N  = Na + Nb;                  // 400000  (Da=Db=64 assumed)

    // workspace layout (~7.3 MB)
    char* ws = (char*)d_ws;
    unsigned* bitmap = (unsigned*)ws;                              // 2 MB
    unsigned* wp     = (unsigned*)(ws + (size_t)NWORDS * 4);       // 2 MB
    unsigned* sums   = (unsigned*)(ws + (size_t)NWORDS * 8);       // 2 KB
    unsigned* offs   = sums + NCHUNK;                              // 2 KB
    unsigned* total  = offs + NCHUNK;                              // 4 B
    int*      invA   = (int*)(total + 1);                          // N ints
    int*      invB   = invA + N;                                   // N ints

    int*   outLoc  = (int*)d_out;                     // N x 3 int32 (bit-cast)
    float* outFeat = (float*)d_out + (size_t)N * 3;   // N x 128 f32

    dim3 b256(256);
    k_fill_u32<<<(NWORDS + 255) / 256, b256, 0, stream>>>(bitmap, 0u, (int)NWORDS);
    k_fill_u32<<<(2 * N + 255) / 256, b256, 0, stream>>>((unsigned*)invA, 0xFFFFFFFFu, 2 * N);
    k_bitmap<<<(N + 255) / 256, b256, 0, stream>>>(aloc, bloc, bitmap, Na, Nb);
    k_chunksums<<<(NCHUNK + 255) / 256, b256, 0, stream>>>(bitmap, sums);
    k_scan<<<1, NCHUNK, 0, stream>>>(sums, offs, total);
    k_wordprefix<<<(NCHUNK + 255) / 256, b256, 0, stream>>>(bitmap, offs, wp);
    k_inverse<<<(N + 255) / 256, b256, 0, stream>>>(aloc, bloc, bitmap, wp, invA, invB, Na, Nb);
    k_fill_loc<<<(NWORDS + 255) / 256, b256, 0, stream>>>(bitmap, wp, outLoc);
    k_fill_tail<<<(N + 255) / 256, b256, 0, stream>>>(outLoc, total, N);

    const int rowsPerBlock = 64;             // 8 waves x 8 rows
    k_gather_features<<<(N + rowsPerBlock - 1) / rowsPerBlock, b256, 0, stream>>>(
        afeat, bfeat, invA, invB, outFeat, N, Na, Nb);
}